// Warp_Function_893353197620
// MI455X (gfx1250) — compile-verified
//
#include <hip/hip_runtime.h>
#include <hip/hip_bf16.h>

// Bilinear flow-warp: out[b,c,h,w] = bilinear_sample(img[b,c], (w+flo_x, h+flo_y))
// img [8,16,512,512] f32, flo [8,2,512,512] f32.
//
//  - one 256-thread block == one image row -> (b,h) scalar per block; all
//    uniform address components in SGPRs, gathers via saddr + voffset + imm24
//  - two x-taps per row fetched as ONE global_load_b64 (edge clamping folded
//    into per-pixel weights via flags aL/cL; exact reference arithmetic in
//    all clamp cases)
//  - integer clamp pipeline: one cvt + int min/max instead of float clamps
//  - 2 adjacent pixels per thread -> b64 NT stores / b64 NT flo loads
//  - NT policy on flo/out keeps the 192MB L2 dedicated to the 128MB img set
//  - WMMA intentionally unused: per-pixel weights cannot form a shared WMMA
//    operand; matrix units cannot accelerate a gather op.

#define B_ 8
#define C_ 16
#define H_ 512
#define W_ 512

constexpr int HW  = H_ * W_;       // 262144
constexpr int CHW = C_ * HW;       // 4194304

typedef float f32x2 __attribute__((ext_vector_type(2)));

// 8-byte load with only 4-byte alignment guarantee (gfx1250 VMEM allows it).
__device__ __forceinline__ f32x2 load_pair(const float* p) {
    f32x2 r;
    __builtin_memcpy(&r, p, 8);
    return r;
}

struct PixSetup {
    int   o0, o1;                   // word offsets of (y0,xb) and (y1,xb)
    float wL0, wH0, wL1, wH1;       // folded weights for {lo,hi} x {row0,row1}
};

__device__ __forceinline__ PixSetup setup_pixel(float pos_x, float pos_y) {
    const float x0f = floorf(pos_x);
    const float y0f = floorf(pos_y);
    const float xw  = pos_x - x0f;        // weights from UNclamped floor (as in ref)
    const float yw  = pos_y - y0f;

    // x0f/y0f are integral -> integer clamps; cvt truncation == floor here.
    const int xb = min(max((int)x0f, 0), W_ - 2);     // lo column of b64 row load
    const int y0 = min(max((int)y0f, 0), H_ - 1);
    const int y1 = min(max(y0 + 1,  0), H_ - 1);

    // Tap routing flags (x0f integral):
    //   aL: tap x0 -> lo word  (x0f < 511  <=>  clamp(x0f) == xb)
    //   cL: tap x1 -> lo word  (x0f < 0    <=>  clamp(x0f+1) == 0 == xb)
    const bool aL = (x0f < (float)(W_ - 1));
    const bool cL = (x0f < 0.f);

    const float wa = (1.f - xw) * (1.f - yw);
    const float wb = (1.f - xw) * yw;
    const float wc = xw * (1.f - yw);
    const float wd = xw * yw;

    PixSetup s;
    s.o0  = y0 * W_ + xb;
    s.o1  = y1 * W_ + xb;
    s.wL0 = (aL ? wa : 0.f) + (cL ? wc : 0.f);
    s.wH0 = (aL ? 0.f : wa) + (cL ? 0.f : wc);
    s.wL1 = (aL ? wb : 0.f) + (cL ? wd : 0.f);
    s.wH1 = (aL ? 0.f : wb) + (cL ? 0.f : wd);
    return s;
}

__global__ __launch_bounds__(256) void warp_bilinear_kernel(
    const float* __restrict__ img,
    const float* __restrict__ flo,
    float* __restrict__ out)
{
    // One block per image row: blockIdx.x in [0, B*H); scalar (b,h) per block.
    const int bh = blockIdx.x;
    const int h  = bh & (H_ - 1);
    const int b  = bh >> 9;
    const int w  = (int)threadIdx.x * 2;          // per-lane even column

    // Scalar (SGPR) bases; per-lane offset is just w words.
    const float* floB = flo + b * 2 * HW + h * W_;
    const float* imgb = img + b * CHW;            // gathers: saddr base
    float*       outB = out + b * CHW + h * W_;   // stores: saddr base

    // flo read-once -> non-temporal b64 loads (8B aligned: w even)
    const f32x2 fx = __builtin_nontemporal_load((const f32x2*)(floB + w));
    const f32x2 fy = __builtin_nontemporal_load((const f32x2*)(floB + HW + w));

    const float hf = (float)h;
    const PixSetup s0 = setup_pixel((float)w       + fx.x, hf + fy.x);
    const PixSetup s1 = setup_pixel((float)(w + 1) + fx.y, hf + fy.y);

#pragma unroll
    for (int c = 0; c < C_; ++c) {
        // saddr form: scalar base + per-lane voffset (set up once) + imm24/SGPR
        const f32x2 r00 = load_pair(imgb + c * HW + s0.o0);  // pixel0 row y0 {lo,hi}
        const f32x2 r01 = load_pair(imgb + c * HW + s0.o1);  // pixel0 row y1
        const f32x2 r10 = load_pair(imgb + c * HW + s1.o0);  // pixel1 row y0
        const f32x2 r11 = load_pair(imgb + c * HW + s1.o1);  // pixel1 row y1

        f32x2 v;
        v.x = fmaf(s0.wL0, r00.x, fmaf(s0.wH0, r00.y,
              fmaf(s0.wL1, r01.x, s0.wH1 * r01.y)));
        v.y = fmaf(s1.wL0, r10.x, fmaf(s1.wH0, r10.y,
              fmaf(s1.wL1, r11.x, s1.wH1 * r11.y)));

        // write-once stream -> non-temporal b64 store (keep L2 for img)
        __builtin_nontemporal_store(v, (f32x2*)(outB + c * HW + w));
    }
}

extern "C" void kernel_launch(void* const* d_in, const int* in_sizes, int n_in,
                              void* d_out, int out_size, void* d_ws, size_t ws_size,
                              hipStream_t stream)
{
    const float* img = (const float*)d_in[0];   // [8,16,512,512] f32
    const float* flo = (const float*)d_in[1];   // [8, 2,512,512] f32
    float* out = (float*)d_out;                 // [8,16,512,512] f32

    const int grid  = B_ * H_;                  // one block per (b,h) row: 4096
    const int block = W_ / 2;                   // 256 threads = 8 wave32 waves
    warp_bilinear_kernel<<<grid, block, 0, stream>>>(img, flo, out);
}